// HP_PPI_Prediction_Model_3908420239628
// MI455X (gfx1250) — compile-verified
//
#include <hip/hip_runtime.h>
#include <hip/hip_bf16.h>

typedef __attribute__((ext_vector_type(2))) float v2f;
typedef __attribute__((ext_vector_type(8))) float v8f;

#define H 128
#define D_EDGE 16
#define N_MOUSE 50000
#define N_VIRUS 10000
#define NEDGE   500000
#define NPAIR   100000
#define D_IN    1024

// ---------------------------------------------------------------------------
// GEMM: C[M x 128] = A[M x K] @ B[K x 128] + bias  (f32, WMMA 16x16x4)
// block = 256 threads (8 waves). Each block: 16 rows. wave w -> cols [16w,16w+16).
// M must be a multiple of 16, K a multiple of 128.
// ---------------------------------------------------------------------------
__global__ void gemm_wmma_bias(const float* __restrict__ A,
                               const float* __restrict__ B,
                               const float* __restrict__ bias,
                               float* __restrict__ C,
                               int M, int K)
{
    __shared__ __align__(16) float As[16 * 132];   // padded stride 132

    const int tid  = threadIdx.x;
    const int wave = tid >> 5;
    const int lane = tid & 31;
    const int hi   = lane >> 4;      // 0: lanes 0-15, 1: lanes 16-31
    const int ml   = lane & 15;      // row (A) / col (B,C) within tile
    const int row0 = blockIdx.x * 16;
    const int n0   = wave * 16;

    v8f acc = {};

    for (int k0 = 0; k0 < K; k0 += 128) {
        // stage A[16][128] chunk into LDS: 2048 floats / 256 thr = 2x float4
        #pragma unroll
        for (int it = 0; it < 2; ++it) {
            int i   = tid * 4 + it * 1024;
            int row = i >> 7;
            int col = i & 127;
            float4 val = *(const float4*)(A + (size_t)(row0 + row) * K + k0 + col);
            *(float4*)(&As[row * 132 + col]) = val;
        }
        __syncthreads();

        #pragma unroll 8
        for (int kk = 0; kk < 128; kk += 4) {
            // A frag: lane<16 -> K=kk,kk+1 ; lane>=16 -> K=kk+2,kk+3 (ISA 7.12.2)
            v2f a = *(const v2f*)(&As[ml * 132 + kk + 2 * hi]);
            // B frag: b.x = B[kk+2*hi][n], b.y = B[kk+2*hi+1][n]
            const float* bp = B + (size_t)(k0 + kk + 2 * hi) * H + n0 + ml;
            v2f b;
            b.x = bp[0];
            b.y = bp[H];
            acc = __builtin_amdgcn_wmma_f32_16x16x4_f32(
                      false, a, false, b, (short)0, acc, false, false);
        }
        __syncthreads();
    }

    const float bval = bias ? bias[n0 + ml] : 0.0f;
    // C/D layout: VGPR r -> row r (lanes 0-15) / row r+8 (lanes 16-31), col = ml
    #pragma unroll
    for (int r = 0; r < 8; ++r) {
        int m = row0 + r + 8 * hi;
        C[(size_t)m * H + n0 + ml] = acc[r] + bval;
    }
}

// ---------------------------------------------------------------------------
// Edge kernel: fused edge-projection + gate + scatter-add.
// out_dst must already contain skip(x_dst) + bias_conv.
// 256 threads = 2 edges x 128 channels.
// ---------------------------------------------------------------------------
__global__ void edge_conv(const float* __restrict__ kbuf,
                          const float* __restrict__ qbuf,
                          const float* __restrict__ vbuf,
                          const float* __restrict__ xe,
                          const float* __restrict__ We,
                          const float* __restrict__ be,
                          const int*   __restrict__ src,
                          const int*   __restrict__ dst,
                          float*       __restrict__ out_dst,
                          int E)
{
    const int e = blockIdx.x * 2 + (threadIdx.x >> 7);
    const int h = threadIdx.x & 127;
    if (e >= E) return;

    const int s = src[e];
    const int d = dst[e];

    // ea[h] = be[h] + sum_d xe[e][d] * We[d][h]   (16-wide K, fused)
    float ea = be[h];
    const float* xrow = xe + (size_t)e * D_EDGE;
    #pragma unroll
    for (int dd = 0; dd < D_EDGE; ++dd)
        ea = fmaf(xrow[dd], We[dd * H + h], ea);

    const float ki = kbuf[(size_t)d * H + h] + ea;
    const float qj = qbuf[(size_t)s * H + h] + ea;
    const float vj = vbuf[(size_t)s * H + h] + ea;

    const float g = 1.0f / (1.0f + __expf(-(ki + qj)));
    atomicAdd(&out_dst[(size_t)d * H + h], g * vj);
}

__global__ void relu_inplace(float* __restrict__ x, int n)
{
    int i = blockIdx.x * blockDim.x + threadIdx.x;
    if (i < n) x[i] = fmaxf(x[i], 0.0f);
}

// ---------------------------------------------------------------------------
// Classifier: out[p] = concat(hm[im[p]], hv[iv[p]]) @ Wc[256x2] + bc
// one wave per pair; shuffle reduction.
// ---------------------------------------------------------------------------
__global__ void classifier(const float* __restrict__ hm,
                           const float* __restrict__ hv,
                           const int*   __restrict__ im,
                           const int*   __restrict__ iv,
                           const float* __restrict__ Wc,
                           const float* __restrict__ bc,
                           float*       __restrict__ out,
                           int P)
{
    const int p    = blockIdx.x * 8 + (threadIdx.x >> 5);
    const int lane = threadIdx.x & 31;
    if (p >= P) return;

    const float4 am = ((const float4*)(hm + (size_t)im[p] * H))[lane];
    const float4 av = ((const float4*)(hv + (size_t)iv[p] * H))[lane];

    float c0 = 0.f, c1 = 0.f;
    const int ch = lane * 4;
    const float a[4] = {am.x, am.y, am.z, am.w};
    const float b[4] = {av.x, av.y, av.z, av.w};
    #pragma unroll
    for (int i = 0; i < 4; ++i) {
        c0 = fmaf(a[i], Wc[(ch + i) * 2 + 0], c0);
        c1 = fmaf(a[i], Wc[(ch + i) * 2 + 1], c1);
        c0 = fmaf(b[i], Wc[(H + ch + i) * 2 + 0], c0);
        c1 = fmaf(b[i], Wc[(H + ch + i) * 2 + 1], c1);
    }
    #pragma unroll
    for (int off = 16; off > 0; off >>= 1) {
        c0 += __shfl_down(c0, off, 32);
        c1 += __shfl_down(c1, off, 32);
    }
    if (lane == 0) {
        out[(size_t)p * 2 + 0] = c0 + bc[0];
        out[(size_t)p * 2 + 1] = c1 + bc[1];
    }
}

// ---------------------------------------------------------------------------
// host side
// ---------------------------------------------------------------------------
static inline void gemm(const float* A, const float* B, const float* bias,
                        float* C, int M, int K, hipStream_t s)
{
    gemm_wmma_bias<<<dim3(M / 16), dim3(256), 0, s>>>(A, B, bias, C, M, K);
}

extern "C" void kernel_launch(void* const* d_in, const int* in_sizes, int n_in,
                              void* d_out, int out_size, void* d_ws, size_t ws_size,
                              hipStream_t stream)
{
    const float* x_mouse  = (const float*)d_in[0];
    const float* x_virus  = (const float*)d_in[1];
    const float* xe_mv    = (const float*)d_in[2];
    const float* xe_vm    = (const float*)d_in[3];
    const float* W_mouse  = (const float*)d_in[4];
    const float* b_mouse  = (const float*)d_in[5];
    const float* W_virus  = (const float*)d_in[6];
    const float* b_virus  = (const float*)d_in[7];
    const float* W_key    = (const float*)d_in[8];
    const float* b_key    = (const float*)d_in[9];
    const float* W_query  = (const float*)d_in[10];
    const float* b_query  = (const float*)d_in[11];
    const float* W_value  = (const float*)d_in[12];
    const float* b_value  = (const float*)d_in[13];
    const float* W_edge   = (const float*)d_in[14];
    const float* b_edge   = (const float*)d_in[15];
    const float* W_skip   = (const float*)d_in[16];
    const float* bias_cv  = (const float*)d_in[17];
    const float* W_cls    = (const float*)d_in[18];
    const float* b_cls    = (const float*)d_in[19];
    const int* ei_mv_src  = (const int*)d_in[20];
    const int* ei_mv_dst  = (const int*)d_in[21];
    const int* ei_vm_src  = (const int*)d_in[22];
    const int* ei_vm_dst  = (const int*)d_in[23];
    const int* eli_mouse  = (const int*)d_in[24];
    const int* eli_virus  = (const int*)d_in[25];
    float* out = (float*)d_out;

    // workspace layout (floats)
    float* ws  = (float*)d_ws;
    const size_t SZ_M = (size_t)N_MOUSE * H;   // 6.4M
    const size_t SZ_V = (size_t)N_VIRUS * H;   // 1.28M
    float* hm0 = ws;
    float* hm1 = hm0 + SZ_M;
    float* hv0 = hm1 + SZ_M;
    float* hv1 = hv0 + SZ_V;
    float* kb  = hv1 + SZ_V;
    float* qb  = kb + SZ_M;
    float* vb  = qb + SZ_M;

    const int relu_m_blocks = (int)((SZ_M + 255) / 256);
    const int relu_v_blocks = (int)((SZ_V + 255) / 256);
    const int edge_blocks   = (NEDGE + 1) / 2;

    // --- input projections -------------------------------------------------
    gemm(x_mouse, W_mouse, b_mouse, hm0, N_MOUSE, D_IN, stream);
    gemm(x_virus, W_virus, b_virus, hv0, N_VIRUS, D_IN, stream);

    // --- two layers, ping-pong (hm0/hv0 -> hm1/hv1 -> hm0/hv0) ------------
    float* hm_in = hm0; float* hv_in = hv0;
    float* hm_out = hm1; float* hv_out = hv1;

    for (int layer = 0; layer < 2; ++layer) {
        const int i_mv = layer * 2;        // mouse -> virus conv
        const int i_vm = layer * 2 + 1;    // virus -> mouse conv

        // conv mv: dst = virus
        {
            const int i = i_mv;
            const float* Wk = W_key   + (size_t)i * H * H;
            const float* Wq = W_query + (size_t)i * H * H;
            const float* Wv = W_value + (size_t)i * H * H;
            const float* We = W_edge  + (size_t)i * D_EDGE * H;
            const float* Ws = W_skip  + (size_t)i * H * H;
            gemm(hv_in, Ws, bias_cv + (size_t)i * H, hv_out, N_VIRUS, H, stream);
            gemm(hv_in, Wk, b_key   + (size_t)i * H, kb, N_VIRUS, H, stream);
            gemm(hm_in, Wq, b_query + (size_t)i * H, qb, N_MOUSE, H, stream);
            gemm(hm_in, Wv, b_value + (size_t)i * H, vb, N_MOUSE, H, stream);
            edge_conv<<<dim3(edge_blocks), dim3(256), 0, stream>>>(
                kb, qb, vb, xe_mv, We, b_edge + (size_t)i * H,
                ei_mv_src, ei_mv_dst, hv_out, NEDGE);
        }
        // conv vm: dst = mouse
        {
            const int i = i_vm;
            const float* Wk = W_key   + (size_t)i * H * H;
            const float* Wq = W_query + (size_t)i * H * H;
            const float* Wv = W_value + (size_t)i * H * H;
            const float* We = W_edge  + (size_t)i * D_EDGE * H;
            const float* Ws = W_skip  + (size_t)i * H * H;
            gemm(hm_in, Ws, bias_cv + (size_t)i * H, hm_out, N_MOUSE, H, stream);
            gemm(hm_in, Wk, b_key   + (size_t)i * H, kb, N_MOUSE, H, stream);
            gemm(hv_in, Wq, b_query + (size_t)i * H, qb, N_VIRUS, H, stream);
            gemm(hv_in, Wv, b_value + (size_t)i * H, vb, N_VIRUS, H, stream);
            edge_conv<<<dim3(edge_blocks), dim3(256), 0, stream>>>(
                kb, qb, vb, xe_vm, We, b_edge + (size_t)i * H,
                ei_vm_src, ei_vm_dst, hm_out, NEDGE);
        }
        relu_inplace<<<dim3(relu_v_blocks), dim3(256), 0, stream>>>(hv_out, (int)SZ_V);
        relu_inplace<<<dim3(relu_m_blocks), dim3(256), 0, stream>>>(hm_out, (int)SZ_M);

        // swap ping-pong
        float* t;
        t = hm_in; hm_in = hm_out; hm_out = t;
        t = hv_in; hv_in = hv_out; hv_out = t;
    }

    // --- classifier ---------------------------------------------------------
    classifier<<<dim3((NPAIR + 7) / 8), dim3(256), 0, stream>>>(
        hm_in, hv_in, eli_mouse, eli_virus, W_cls, b_cls, out, NPAIR);
}